// GNN_Disentangle_86191403696584
// MI455X (gfx1250) — compile-verified
//
#include <hip/hip_runtime.h>

typedef __attribute__((ext_vector_type(2))) float v2f;
typedef __attribute__((ext_vector_type(8))) float v8f;

#define HDIM 64

// ---------------------------------------------------------------------------
// Embedding renorm: one wave per gene row; L2 norm via shfl butterfly.
// ---------------------------------------------------------------------------
__global__ void k_emb_norm(const float* __restrict__ T, float* __restrict__ out, int ng) {
    int g = blockIdx.x;
    if (g >= ng) return;
    int lane = threadIdx.x; // blockDim = 32
    float v0 = T[(size_t)g * HDIM + lane];
    float v1 = T[(size_t)g * HDIM + lane + 32];
    float s = v0 * v0 + v1 * v1;
#pragma unroll
    for (int m = 16; m >= 1; m >>= 1) s += __shfl_xor(s, m, 32);
    float nrm = sqrtf(s);
    float scl = fminf(1.0f, 1.0f / fmaxf(nrm, 1e-7f));
    out[(size_t)g * HDIM + lane] = v0 * scl;
    out[(size_t)g * HDIM + lane + 32] = v1 * scl;
}

// ---------------------------------------------------------------------------
// pert_emb / base: rank-1 per-node features.
// ---------------------------------------------------------------------------
__global__ void k_node_scal(const float* __restrict__ x,
                            const float* __restrict__ pW, const float* __restrict__ pb,
                            const float* __restrict__ bW, const float* __restrict__ bb,
                            float* __restrict__ pert, float* __restrict__ base,
                            long long total) {
    long long idx = (long long)blockIdx.x * blockDim.x + threadIdx.x;
    if (idx >= total) return;
    long long n = idx >> 6;
    int f = (int)(idx & 63);
    float x0 = x[2 * n];
    float x1 = x[2 * n + 1];
    pert[idx] = x1 * pW[f] + pb[f];
    base[idx] = x0 * bW[f] + bb[f];
}

// ---------------------------------------------------------------------------
// Degree (self-loop folded into init), then dinv = rsqrt(deg).
// ---------------------------------------------------------------------------
__global__ void k_deg_init(float* __restrict__ deg, int n) {
    long long i = (long long)blockIdx.x * blockDim.x + threadIdx.x;
    if (i < n) deg[i] = 1.0f; // self-loop contribution
}
__global__ void k_deg_edges(const int* __restrict__ dst, float* __restrict__ deg, int e) {
    long long i = (long long)blockIdx.x * blockDim.x + threadIdx.x;
    if (i < e) atomicAdd(&deg[dst[i]], 1.0f);
}
__global__ void k_dinv(const float* __restrict__ deg, float* __restrict__ dinv, int n) {
    long long i = (long long)blockIdx.x * blockDim.x + threadIdx.x;
    if (i < n) {
        float d = deg[i];
        dinv[i] = (d > 0.0f) ? rsqrtf(fmaxf(d, 1.0f)) : 0.0f;
    }
}

// ---------------------------------------------------------------------------
// GCN aggregation: out initialized with the self-loop term (no atomics),
// then one wave per edge does 64 coalesced float atomics.
// ---------------------------------------------------------------------------
__global__ void k_scatter_init(const float* __restrict__ dinv, const float* __restrict__ m,
                               float* __restrict__ out, long long total) {
    long long idx = (long long)blockIdx.x * blockDim.x + threadIdx.x;
    if (idx >= total) return;
    long long n = idx >> 6;
    float di = dinv[n];
    out[idx] = di * di * m[idx];
}
__global__ void k_scatter_edges(const int* __restrict__ src, const int* __restrict__ dst,
                                const float* __restrict__ dinv, const float* __restrict__ m,
                                float* __restrict__ out, int e) {
    long long gid = (long long)blockIdx.x * blockDim.x + threadIdx.x;
    int edge = (int)(gid >> 5);
    int lane = (int)(gid & 31);
    if (edge >= e) return;
    int s = src[edge], d = dst[edge];
    float w = dinv[s] * dinv[d];
    atomicAdd(&out[(long long)d * HDIM + lane], w * m[(long long)s * HDIM + lane]);
    atomicAdd(&out[(long long)d * HDIM + lane + 32], w * m[(long long)s * HDIM + lane + 32]);
}
__global__ void k_bias_relu(float* __restrict__ z, const float* __restrict__ b, long long total) {
    long long idx = (long long)blockIdx.x * blockDim.x + threadIdx.x;
    if (idx >= total) return;
    z[idx] = fmaxf(z[idx] + b[idx & 63], 0.0f);
}

// ---------------------------------------------------------------------------
// Generic WMMA f32 GEMM:  out[n, :NOUT] = concat(A1[n'],A2[n]) @ W^T (+ bias)
// W is [NOUT][K1+K2] row-major. A1 optionally indexed modulo a1mod (embedding
// tile). Block = 128 threads = 4 waves, 64 rows per block; A tile staged in
// LDS (coalesced); each wave owns a 16-row strip, K stepped by 4 per
// v_wmma_f32_16x16x4_f32.
// ---------------------------------------------------------------------------
template <int K1, int K2, int NOUT>
__global__ void __launch_bounds__(128)
k_gemm_wmma(const float* __restrict__ A1, int a1mod,
            const float* __restrict__ A2,
            const float* __restrict__ W,
            const float* __restrict__ bias, // may be null
            float* __restrict__ out, long long nrows) {
    constexpr int KT = K1 + K2;
    constexpr int LDK = KT + 4; // LDS pad against bank conflicts
    __shared__ float sA[64 * LDK];

    const int tid = threadIdx.x;
    const int lane = tid & 31;
    const int wave = tid >> 5;
    const long long rowBase = (long long)blockIdx.x * 64;

    // Stage 64 x KT tile of the (concatenated) A operand into LDS, coalesced.
    for (int idx = tid; idx < 64 * KT; idx += 128) {
        int r = idx / KT;
        int c = idx - r * KT;
        long long g = rowBase + r;
        if (g >= nrows) g = nrows - 1;
        float v;
        if (c < K1) {
            long long rr = a1mod ? (g % a1mod) : g;
            v = A1[rr * K1 + c];
        } else {
            v = A2[g * K2 + (c - K1)];
        }
        sA[r * LDK + c] = v;
    }
    __syncthreads();

    const int m0 = wave * 16;
    const int arow = m0 + (lane & 15);     // A fragment row within tile
    const int koff = (lane >> 4) * 2;      // lanes 16..31 carry K+2,K+3
    const int rbase = m0 + (lane >> 4) * 8;

#pragma unroll
    for (int nt = 0; nt < NOUT / 16; ++nt) {
        v8f acc = {};
        const int ncol = nt * 16 + (lane & 15);
        const float* wrow = W + (long long)ncol * KT;
        for (int k = 0; k < KT; k += 4) {
            v2f a, b;
            a.x = sA[arow * LDK + k + koff];
            a.y = sA[arow * LDK + k + koff + 1];
            b.x = wrow[k + koff];
            b.y = wrow[k + koff + 1];
            acc = __builtin_amdgcn_wmma_f32_16x16x4_f32(
                false, a, false, b, (short)0, acc, false, false);
        }
        float bv = bias ? bias[ncol] : 0.0f;
#pragma unroll
        for (int r = 0; r < 8; ++r) {
            long long row = rowBase + rbase + r;
            if (row < nrows) out[row * NOUT + ncol] = acc[r] + bv;
        }
    }
}

// ---------------------------------------------------------------------------
// BatchNorm: partial sums per block -> float atomics; finalize; fused apply.
// ---------------------------------------------------------------------------
__global__ void k_zero(float* __restrict__ p, int n) {
    int i = blockIdx.x * blockDim.x + threadIdx.x;
    if (i < n) p[i] = 0.0f;
}
template <int F>
__global__ void k_bn_stats(const float* __restrict__ z, float* __restrict__ stats,
                           long long nrows, int rowsPerBlock) {
    int f = threadIdx.x; // blockDim = F
    long long r0 = (long long)blockIdx.x * rowsPerBlock;
    long long r1 = r0 + rowsPerBlock;
    if (r1 > nrows) r1 = nrows;
    float s = 0.0f, sq = 0.0f;
    for (long long r = r0; r < r1; ++r) {
        float v = z[r * F + f];
        s += v;
        sq += v * v;
    }
    atomicAdd(&stats[f], s);
    atomicAdd(&stats[F + f], sq);
}
__global__ void k_bn_finalize(const float* __restrict__ stats,
                              const float* __restrict__ g, const float* __restrict__ be,
                              float* __restrict__ scsh, int F, float invN) {
    int f = blockIdx.x * blockDim.x + threadIdx.x;
    if (f >= F) return;
    float mean = stats[f] * invN;
    float var = stats[F + f] * invN - mean * mean;
    float sc = g[f] * rsqrtf(var + 1e-5f);
    scsh[f] = sc;
    scsh[F + f] = be[f] - mean * sc;
}
template <int F>
__global__ void k_bn_apply_relu(float* __restrict__ z, const float* __restrict__ scsh,
                                long long total) {
    long long idx = (long long)blockIdx.x * blockDim.x + threadIdx.x;
    if (idx >= total) return;
    int f = (int)(idx % F);
    z[idx] = fmaxf(z[idx] * scsh[f] + scsh[F + f], 0.0f);
}

// ---------------------------------------------------------------------------
// Final 64 -> 1 projection: one wave per node, shfl reduction.
// ---------------------------------------------------------------------------
__global__ void k_final(const float* __restrict__ z, const float* __restrict__ w3,
                        const float* __restrict__ b3, float* __restrict__ out, int n) {
    long long gid = (long long)blockIdx.x * blockDim.x + threadIdx.x;
    int node = (int)(gid >> 5);
    int lane = (int)(gid & 31);
    if (node >= n) return;
    float s = z[(long long)node * HDIM + lane] * w3[lane] +
              z[(long long)node * HDIM + lane + 32] * w3[lane + 32];
#pragma unroll
    for (int m = 16; m >= 1; m >>= 1) s += __shfl_xor(s, m, 32);
    if (lane == 0) out[node] = s + b3[0];
}

// ---------------------------------------------------------------------------
extern "C" void kernel_launch(void* const* d_in, const int* in_sizes, int n_in,
                              void* d_out, int out_size, void* d_ws, size_t ws_size,
                              hipStream_t stream) {
    const float* x        = (const float*)d_in[0];
    const int*   ei       = (const int*)d_in[1];
    const float* pert_W   = (const float*)d_in[2];
    const float* pert_b   = (const float*)d_in[3];
    const float* basal_W  = (const float*)d_in[4];
    const float* basal_b  = (const float*)d_in[5];
    const float* emb_tab  = (const float*)d_in[6];
    const float* et_W     = (const float*)d_in[7];
    const float* et_b     = (const float*)d_in[8];
    const float* pbt_W    = (const float*)d_in[9];
    const float* pbt_b    = (const float*)d_in[10];
    const float* gcn1_W   = (const float*)d_in[11];
    const float* gcn1_b   = (const float*)d_in[12];
    const float* gcn2_W   = (const float*)d_in[13];
    const float* gcn2_b   = (const float*)d_in[14];
    const float* rec_W1   = (const float*)d_in[15];
    const float* rec_b1   = (const float*)d_in[16];
    const float* rec_g1   = (const float*)d_in[17];
    const float* rec_be1  = (const float*)d_in[18];
    const float* rec_W2   = (const float*)d_in[19];
    const float* rec_b2   = (const float*)d_in[20];
    const float* rec_g2   = (const float*)d_in[21];
    const float* rec_be2  = (const float*)d_in[22];
    const float* rec_W3   = (const float*)d_in[23];
    const float* rec_b3   = (const float*)d_in[24];

    const int N  = in_sizes[0] / 2;
    const int E  = in_sizes[1] / 2;
    const int NG = in_sizes[6] / HDIM;
    const int*   src = ei;
    const int*   dst = ei + E;
    const long long NH = (long long)N * HDIM;

    // ---- workspace layout (floats), ~167 MB ----
    float* ws = (float*)d_ws;
    size_t o = 0;
    float* emb_n    = ws + o; o += (size_t)NG * HDIM;
    float* deg      = ws + o; o += (size_t)N;
    float* dinv     = ws + o; o += (size_t)N;
    float* base_emb = ws + o; o += (size_t)N * HDIM;   // later reused as z2
    float* buf2     = ws + o; o += (size_t)2 * N * HDIM; // bufA|bufB, also z1 [N][128]
    float* bufA     = buf2;
    float* bufB     = buf2 + (size_t)N * HDIM;
    float* hbuf     = ws + o; o += (size_t)N * HDIM;
    float* stats    = ws + o; o += 256;
    float* scsh     = ws + o; o += 256;
    (void)ws_size; (void)n_in; (void)out_size;

    const int T = 256;
    const int nhBlocks   = (int)((NH + T - 1) / T);
    const int gemmBlocks = (N + 63) / 64;
    const int edgeBlocks = (E + T - 1) / T;
    const int nBlocks    = (N + T - 1) / T;
    const int scatBlocks = (int)(((long long)E * 32 + T - 1) / T);
    const int waveBlocks = (int)(((long long)N * 32 + T - 1) / T);
    const int bnBlocks   = (N + 255) / 256;
    const float invN = 1.0f / (float)N;

    // ---- embedding renorm + rank-1 node features + GCN norm ----
    k_emb_norm<<<NG, 32, 0, stream>>>(emb_tab, emb_n, NG);
    k_node_scal<<<nhBlocks, T, 0, stream>>>(x, pert_W, pert_b, basal_W, basal_b,
                                            bufA /*pert*/, bufB /*base*/, NH);
    k_deg_init<<<nBlocks, T, 0, stream>>>(deg, N);
    k_deg_edges<<<edgeBlocks, T, 0, stream>>>(dst, deg, E);
    k_dinv<<<nBlocks, T, 0, stream>>>(deg, dinv, N);

    // ---- base_emb = concat(emb_n[n % NG], base) @ emb_trans_W^T + b ----
    k_gemm_wmma<64, 64, 64><<<gemmBlocks, 128, 0, stream>>>(
        emb_n, NG, bufB, et_W, et_b, base_emb, N);
    // ---- h = pbt(pert_emb, base_emb) ----
    k_gemm_wmma<64, 64, 64><<<gemmBlocks, 128, 0, stream>>>(
        bufA, 0, base_emb, pbt_W, pbt_b, hbuf, N);

    // ---- two GCN layers ----
    const float* gcnW[2] = {gcn1_W, gcn2_W};
    const float* gcnB[2] = {gcn1_b, gcn2_b};
    for (int l = 0; l < 2; ++l) {
        // m = h @ W^T   (bias applied after aggregation, per PyG)
        k_gemm_wmma<64, 0, 64><<<gemmBlocks, 128, 0, stream>>>(
            hbuf, 0, (const float*)nullptr, gcnW[l], (const float*)nullptr, bufA, N);
        // out = dinv^2 * m (self loops) + scatter(norm * m[src] -> dst)
        k_scatter_init<<<nhBlocks, T, 0, stream>>>(dinv, bufA, bufB, NH);
        k_scatter_edges<<<scatBlocks, T, 0, stream>>>(src, dst, dinv, bufA, bufB, E);
        k_bias_relu<<<nhBlocks, T, 0, stream>>>(bufB, gcnB[l], NH);
        // h = pbt(relu(agg), base_emb)
        k_gemm_wmma<64, 64, 64><<<gemmBlocks, 128, 0, stream>>>(
            bufB, 0, base_emb, pbt_W, pbt_b, hbuf, N);
    }

    // ---- recovery MLP ----
    float* z1 = buf2; // [N][128], bufA/bufB region is contiguous
    k_gemm_wmma<64, 0, 128><<<gemmBlocks, 128, 0, stream>>>(
        hbuf, 0, (const float*)nullptr, rec_W1, rec_b1, z1, N);

    k_zero<<<1, 256, 0, stream>>>(stats, 256);
    k_bn_stats<128><<<bnBlocks, 128, 0, stream>>>(z1, stats, N, 256);
    k_bn_finalize<<<1, 128, 0, stream>>>(stats, rec_g1, rec_be1, scsh, 128, invN);
    k_bn_apply_relu<128><<<(int)(((long long)N * 128 + T - 1) / T), T, 0, stream>>>(
        z1, scsh, (long long)N * 128);

    float* z2 = base_emb; // base_emb no longer needed
    k_gemm_wmma<128, 0, 64><<<gemmBlocks, 128, 0, stream>>>(
        z1, 0, (const float*)nullptr, rec_W2, rec_b2, z2, N);

    k_zero<<<1, 256, 0, stream>>>(stats, 256);
    k_bn_stats<64><<<bnBlocks, 64, 0, stream>>>(z2, stats, N, 256);
    k_bn_finalize<<<1, 64, 0, stream>>>(stats, rec_g2, rec_be2, scsh, 64, invN);
    k_bn_apply_relu<64><<<nhBlocks, T, 0, stream>>>(z2, scsh, NH);

    // ---- out[n] = z2[n] . rec_W3 + b3 ----
    k_final<<<waveBlocks, T, 0, stream>>>(z2, rec_W3, rec_b3, (float*)d_out, N);
}